// Transformer_first_146028888654
// MI455X (gfx1250) — compile-verified
//
#include <hip/hip_runtime.h>
#include <hip/hip_bf16.h>

// ---------------------------------------------------------------------------
// Problem constants (from reference)
// ---------------------------------------------------------------------------
#define TV 4      // views
#define TB 4      // batch
#define TN 1024   // tokens
#define TC 256    // channels
#define TH 8      // heads
#define THD 32    // head dim
#define THID 1024 // mlp hidden
#define TSCALE 0.17677669529663687f  // 32^-0.5
#define KC 16     // GEMM K-chunk staged in LDS

typedef __attribute__((ext_vector_type(2))) float v2f;
typedef __attribute__((ext_vector_type(8))) float v8f;
typedef __attribute__((vector_size(16))) int v4i;   // matches async-lds builtin param

// fp32 tensor-core MAC: D(16x16,f32) = A(16x4,f32) * B(4x16,f32) + C
__device__ __forceinline__ v8f wmma_f32(v2f a, v2f b, v8f c) {
  return __builtin_amdgcn_wmma_f32_16x16x4_f32(
      /*neg_a=*/false, a, /*neg_b=*/false, b,
      /*c_mod=*/(short)0, c, /*reuse_a=*/false, /*reuse_b=*/false);
}

// ---- CDNA5 async global->LDS copy (ASYNCcnt-tracked DMA), 16B per lane ----
__device__ __forceinline__ void async_copy_b128(const float* src, float* ldsDst) {
#if __has_builtin(__builtin_amdgcn_global_load_async_to_lds_b128)
  __builtin_amdgcn_global_load_async_to_lds_b128(
      (__attribute__((address_space(1))) v4i*)(float*)src,
      (__attribute__((address_space(3))) v4i*)ldsDst, 0, 0);
#else
  unsigned lds_off = (unsigned)(unsigned long long)ldsDst;  // LDS addr = low 32 bits
  unsigned long long gaddr = (unsigned long long)src;
  asm volatile("global_load_async_to_lds_b128 %0, %1, off"
               :: "v"(lds_off), "v"(gaddr) : "memory");
#endif
}

__device__ __forceinline__ void wait_async_le1() {
#if __has_builtin(__builtin_amdgcn_s_wait_asynccnt)
  __builtin_amdgcn_s_wait_asynccnt(1);
#else
  asm volatile("s_wait_asynccnt 0x1" ::: "memory");
#endif
}
__device__ __forceinline__ void wait_async_0() {
#if __has_builtin(__builtin_amdgcn_s_wait_asynccnt)
  __builtin_amdgcn_s_wait_asynccnt(0);
#else
  asm volatile("s_wait_asynccnt 0x0" ::: "memory");
#endif
}

// ---------------------------------------------------------------------------
// patches + pos_emb broadcast; also extract view-0 into x
// ---------------------------------------------------------------------------
__global__ __launch_bounds__(256) void k_add_pos(
    const float* __restrict__ patches, const float* __restrict__ pos,
    float* __restrict__ xs, float* __restrict__ x) {
  int i = blockIdx.x * 256 + threadIdx.x;           // over V*B*N*C
  int c = i & (TC - 1);
  int n = (i >> 8) & (TN - 1);
  int vb = i >> 18;                                  // v*B + b
  int v = vb >> 2;
  float val = patches[i] + pos[((size_t)v * TN + n) * TC + c];
  xs[i] = val;
  if (vb < TB) x[i] = val;                           // v == 0 block
}

// ---------------------------------------------------------------------------
// Row LayerNorm over C=256: one wave per row, 8 waves per block
// ---------------------------------------------------------------------------
__global__ __launch_bounds__(256) void k_ln(
    const float* __restrict__ X, const float* __restrict__ g,
    const float* __restrict__ bta, float* __restrict__ Y, int M) {
  int wave = threadIdx.x >> 5;
  int lane = threadIdx.x & 31;
  int row = blockIdx.x * 8 + wave;
  if (row >= M) return;
  const float* xr = X + (size_t)row * TC;
  float vals[8];
  float s = 0.f, s2 = 0.f;
#pragma unroll
  for (int j = 0; j < 8; ++j) {
    float t = xr[lane + 32 * j];
    vals[j] = t; s += t; s2 += t * t;
  }
#pragma unroll
  for (int off = 16; off >= 1; off >>= 1) {
    s  += __shfl_xor(s, off, 32);
    s2 += __shfl_xor(s2, off, 32);
  }
  float m   = s  * (1.f / TC);
  float var = s2 * (1.f / TC) - m * m;
  float inv = rsqrtf(var + 1e-6f);
  float* yr = Y + (size_t)row * TC;
#pragma unroll
  for (int j = 0; j < 8; ++j) {
    int c = lane + 32 * j;
    yr[c] = (vals[j] - m) * inv * g[c] + bta[c];
  }
}

// ---------------------------------------------------------------------------
// GEMM: Y[M,Nout] = epilogue( X[M,K] @ W[K,Nout] + bias ) (+ residual)
// 128-thread block (4 waves) computes a 64x32 tile. The 16x32 B panel for
// each K-chunk is DMA'd into LDS with GLOBAL_LOAD_ASYNC_TO_LDS_B128
// (double-buffered, overlapped with WMMA compute, s_wait_asynccnt paced);
// all 4 waves share it. Each wave runs V_WMMA_F32_16X16X4_F32 on its M-tile.
// ---------------------------------------------------------------------------
__global__ __launch_bounds__(128) void k_gemm(
    const float* __restrict__ X, const float* __restrict__ W,
    const float* __restrict__ bias, const float* __restrict__ res,
    float* __restrict__ Y, int M, int K, int Nout, int do_gelu) {
  __shared__ float Bs[2][KC * 32];

  const int tid  = threadIdx.x;
  const int wave = tid >> 5;
  const int lane = tid & 31;
  const int half = lane >> 4;     // K sub-slot
  const int lidx = lane & 15;     // M (A) / N (B,C) index
  const int row0 = blockIdx.x * 64 + wave * 16;
  const int col0 = blockIdx.y * 32;

  // staging map: thread -> 16 consecutive bytes of the B panel
  const int srow = tid >> 3;        // 0..15 (k within chunk)
  const int scol = (tid & 7) * 4;   // 0,4,...,28 (column within 32-wide panel)

  // prologue: stage chunk 0 into buffer 0
  async_copy_b128(W + (size_t)srow * Nout + col0 + scol, &Bs[0][srow * 32 + scol]);

  v8f acc0 = {};
  v8f acc1 = {};
  const float* Arow = X + (size_t)(row0 + lidx) * K;

  for (int kb = 0; kb < K; kb += KC) {
    const int buf = (kb / KC) & 1;
    const bool more = (kb + KC) < K;
    if (more) {
      // kick off DMA of the next panel into the other buffer
      async_copy_b128(W + (size_t)(kb + KC + srow) * Nout + col0 + scol,
                      &Bs[buf ^ 1][srow * 32 + scol]);
      // keep the A stream warm too (global_prefetch_b8)
      __builtin_prefetch(Arow + kb + KC, 0, 1);
      wait_async_le1();             // current panel done (async loads in order)
    } else {
      wait_async_0();
    }
    __syncthreads();                // panel visible to all 4 waves

#pragma unroll
    for (int k = 0; k < KC; k += 4) {
      float2 at = *reinterpret_cast<const float2*>(Arow + kb + k + half * 2);
      v2f a; a.x = at.x; a.y = at.y;
      const int kk = k + half * 2;
      v2f b0, b1;
      b0.x = Bs[buf][(kk + 0) * 32 + lidx];
      b0.y = Bs[buf][(kk + 1) * 32 + lidx];
      b1.x = Bs[buf][(kk + 0) * 32 + 16 + lidx];
      b1.y = Bs[buf][(kk + 1) * 32 + 16 + lidx];
      acc0 = wmma_f32(a, b0, acc0);
      acc1 = wmma_f32(a, b1, acc1);
    }
    __syncthreads();                // done reading buf before it is re-staged
  }

#pragma unroll
  for (int r = 0; r < 8; ++r) {
    int row = row0 + r + half * 8;
#pragma unroll
    for (int c = 0; c < 2; ++c) {
      int col = col0 + c * 16 + lidx;
      float v = (c == 0) ? acc0[r] : acc1[r];
      if (bias) v += bias[col];
      if (do_gelu) v = 0.5f * v * (1.f + erff(v * 0.70710678118654752f));
      if (res) v += res[(size_t)row * Nout + col];
      Y[(size_t)row * Nout + col] = v;
    }
  }
}

// ---------------------------------------------------------------------------
// Layer-0 cross-view attention: per (b,n) softmax over V=4 views.
// One wave per token; lanes span head-dim d=0..31.
// ---------------------------------------------------------------------------
__global__ __launch_bounds__(32) void k_attn0(
    const float* __restrict__ Q, const float* __restrict__ KV0,
    float* __restrict__ O) {
  const int lane = threadIdx.x;
  const int bn = blockIdx.x;                  // b*N + n
  const int b = bn >> 10, n = bn & (TN - 1);
  float kvals[TV], vvals[TV];
#pragma unroll
  for (int v = 0; v < TV; ++v) {
    const float* p = KV0 + ((size_t)((v * TB + b) * TN + n)) * (2 * THD);
    kvals[v] = p[lane];
    vvals[v] = p[THD + lane];
  }
  const float* qb = Q + (size_t)bn * TC;
  float* ob = O + (size_t)bn * TC;
#pragma unroll
  for (int h = 0; h < TH; ++h) {
    float qv = qb[h * THD + lane];
    float dots[TV];
#pragma unroll
    for (int v = 0; v < TV; ++v) {
      float d = qv * kvals[v];
#pragma unroll
      for (int off = 16; off >= 1; off >>= 1) d += __shfl_xor(d, off, 32);
      dots[v] = d * TSCALE;
    }
    float mx = fmaxf(fmaxf(dots[0], dots[1]), fmaxf(dots[2], dots[3]));
    float e[TV], sum = 0.f;
#pragma unroll
    for (int v = 0; v < TV; ++v) { e[v] = expf(dots[v] - mx); sum += e[v]; }
    float inv = 1.f / sum;
    float o = 0.f;
#pragma unroll
    for (int v = 0; v < TV; ++v) o += e[v] * inv * vvals[v];
    ob[h * THD + lane] = o;
  }
}

// ---------------------------------------------------------------------------
// Layers 1..3 attention: flash-style, WMMA for QK^T and PV.
// One wave per (b, h, 16-query tile). K/V shared across heads (HD=32).
// ---------------------------------------------------------------------------
__global__ __launch_bounds__(32) void k_attn_flash(
    const float* __restrict__ Q, const float* __restrict__ KV,
    float* __restrict__ O) {
  __shared__ float Ps[16 * 16];                  // P tile for C->A layout flip
  const int lane = threadIdx.x;
  const int half = lane >> 4;
  const int lidx = lane & 15;
  const int bid = blockIdx.x;
  const int it = bid & 63;                       // query tile (N/16 = 64)
  const int h  = (bid >> 6) & (TH - 1);
  const int b  = bid >> 9;

  // preload Q A-fragments for all 8 K-steps (d = 0..31)
  v2f qa[8];
  const float* qrow = Q + ((size_t)(b * TN + it * 16 + lidx)) * TC + h * THD;
#pragma unroll
  for (int kk = 0; kk < 8; ++kk) {
    float2 t = *reinterpret_cast<const float2*>(qrow + kk * 4 + half * 2);
    qa[kk].x = t.x; qa[kk].y = t.y;
  }

  v8f o0 = {}, o1 = {};
  float rm[8], rl[8];
#pragma unroll
  for (int r = 0; r < 8; ++r) { rm[r] = -1e30f; rl[r] = 0.f; }

  for (int jt = 0; jt < TN / 16; ++jt) {
    // ---- S = Q K^T (16x16), K-dim = 32 -> 8 WMMAs --------------------------
    v8f s = {};
    const float* krow = KV + ((size_t)(b * TN + jt * 16 + lidx)) * (2 * THD);
#pragma unroll
    for (int kk = 0; kk < 8; ++kk) {
      float2 t = *reinterpret_cast<const float2*>(krow + kk * 4 + half * 2);
      v2f bk; bk.x = t.x; bk.y = t.y;
      s = wmma_f32(qa[kk], bk, s);
    }
    // ---- online softmax update (rows live in 16-lane halves) --------------
#pragma unroll
    for (int r = 0; r < 8; ++r) {
      float sv = s[r] * TSCALE;
      float mx = sv;
#pragma unroll
      for (int off = 8; off >= 1; off >>= 1) mx = fmaxf(mx, __shfl_xor(mx, off, 32));
      float mnew = fmaxf(rm[r], mx);
      float p = expf(sv - mnew);
      float ps = p;
#pragma unroll
      for (int off = 8; off >= 1; off >>= 1) ps += __shfl_xor(ps, off, 32);
      float al = expf(rm[r] - mnew);
      rl[r] = rl[r] * al + ps;
      rm[r] = mnew;
      o0[r] *= al;
      o1[r] *= al;
      s[r] = p;
    }
    // ---- P through LDS: C-layout -> A-layout -------------------------------
    __syncthreads();
#pragma unroll
    for (int r = 0; r < 8; ++r) Ps[(r + half * 8) * 16 + lidx] = s[r];
    __syncthreads();
    // ---- O += P V (K-dim 16 -> 4 WMMAs per 16-wide d-chunk) ----------------
    const float* vbase = KV + ((size_t)(b * TN + jt * 16)) * (2 * THD) + THD;
#pragma unroll
    for (int kk = 0; kk < 4; ++kk) {
      float2 pt = *reinterpret_cast<const float2*>(&Ps[lidx * 16 + kk * 4 + half * 2]);
      v2f pa; pa.x = pt.x; pa.y = pt.y;
      const int j0 = kk * 4 + half * 2;
      v2f vb0, vb1;
      vb0.x = vbase[(size_t)(j0 + 0) * (2 * THD) + lidx];
      vb0.y = vbase[(size_t)(j0 + 1) * (2 * THD) + lidx];
      vb1.x = vbase[(size_t)(j0 + 0) * (2 * THD) + 16 + lidx];
      vb1.y = vbase[(size_t)(j0 + 1) * (2 * THD) + 16 + lidx];
      o0 = wmma_f32(pa, vb0, o0);
      o1 = wmma_f32(pa, vb1, o1);
    }
  }

#pragma unroll
  for (int r = 0; r < 8; ++r) {
    int row = it * 16 + r + half * 8;
    float inv = 1.f / rl[r];
    float* orow = O + ((size_t)(b * TN + row)) * TC + h * THD;
    orow[lidx]      = o0[r] * inv;
    orow[16 + lidx] = o1[r] * inv;
  }
}

// ---------------------------------------------------------------------------
// Orchestration
// ---------------------------------------------------------------------------
extern "C" void kernel_launch(void* const* d_in, const int* in_sizes, int n_in,
                              void* d_out, int out_size, void* d_ws, size_t ws_size,
                              hipStream_t stream) {
  const float* patches = (const float*)d_in[0];
  const float* pos_emb = (const float*)d_in[1];
  const float* ln1_g = (const float*)d_in[2];
  const float* ln1_b = (const float*)d_in[3];
  const float* ln2_g = (const float*)d_in[4];
  const float* ln2_b = (const float*)d_in[5];
  const float* Wq    = (const float*)d_in[6];
  const float* Wkv   = (const float*)d_in[7];
  const float* projw = (const float*)d_in[8];
  const float* projb = (const float*)d_in[9];
  const float* fc1w  = (const float*)d_in[10];
  const float* fc1b  = (const float*)d_in[11];
  const float* fc2w  = (const float*)d_in[12];
  const float* fc2b  = (const float*)d_in[13];

  float* x = (float*)d_out;                       // running activations (B,N,C)

  float* ws   = (float*)d_ws;
  float* pn_h = ws;                               // 4M fl: pn (layer0) then MLP h
  float* xs   = pn_h + (size_t)TV * TB * TN * TC; // 4M fl: patches+pos
  float* xn   = xs   + (size_t)TV * TB * TN * TC; // 1M fl
  float* q    = xn   + (size_t)TB * TN * TC;      // 1M fl
  float* kv   = q    + (size_t)TB * TN * TC;      // 1M fl (layer0 V*B*N*64)
  float* o    = kv   + (size_t)TV * TB * TN * 2 * THD; // 1M fl

  const int M = TB * TN;                          // 4096 token rows

  k_add_pos<<<(TV * TB * TN * TC) / 256, 256, 0, stream>>>(patches, pos_emb, xs, x);

  for (int i = 0; i < 4; ++i) {
    // xn = LN1(x)
    k_ln<<<M / 8, 256, 0, stream>>>(x, ln1_g, ln1_b, xn, M);
    // q = xn @ Wq[i]
    k_gemm<<<dim3(M / 64, TC / 32), 128, 0, stream>>>(
        xn, Wq + (size_t)i * TC * TC, nullptr, nullptr, q, M, TC, TC, 0);

    if (i == 0) {
      // pn = LN1(all views); kv0 = pn @ Wkv[0]; cross-view attention
      k_ln<<<(TV * M) / 8, 256, 0, stream>>>(xs, ln1_g, ln1_b, pn_h, TV * M);
      k_gemm<<<dim3((TV * M) / 64, (2 * THD) / 32), 128, 0, stream>>>(
          pn_h, Wkv, nullptr, nullptr, kv, TV * M, TC, 2 * THD, 0);
      k_attn0<<<M, 32, 0, stream>>>(q, kv, o);
    } else {
      k_gemm<<<dim3(M / 64, (2 * THD) / 32), 128, 0, stream>>>(
          xn, Wkv + (size_t)i * TC * 2 * THD, nullptr, nullptr, kv, M, TC, 2 * THD, 0);
      k_attn_flash<<<TB * TH * (TN / 16), 32, 0, stream>>>(q, kv, o);
    }

    // x = x + o @ proj_w[i] + proj_b[i]
    k_gemm<<<dim3(M / 64, TC / 32), 128, 0, stream>>>(
        o, projw + (size_t)i * TC * TC, projb + (size_t)i * TC, x, x, M, TC, TC, 0);
    // xn = LN2(x)
    k_ln<<<M / 8, 256, 0, stream>>>(x, ln2_g, ln2_b, xn, M);
    // h = gelu(xn @ fc1 + b1)
    k_gemm<<<dim3(M / 64, THID / 32), 128, 0, stream>>>(
        xn, fc1w + (size_t)i * TC * THID, fc1b + (size_t)i * THID, nullptr,
        pn_h, M, TC, THID, 1);
    // x = x + h @ fc2 + b2
    k_gemm<<<dim3(M / 64, TC / 32), 128, 0, stream>>>(
        pn_h, fc2w + (size_t)i * THID * TC, fc2b + (size_t)i * TC, x, x, M, THID, TC, 0);
  }
}